// GCNEdgeClassifier_18476949307756
// MI455X (gfx1250) — compile-verified
//
#include <hip/hip_runtime.h>

typedef __attribute__((ext_vector_type(2))) float v2f;
typedef __attribute__((ext_vector_type(8))) float v8f;

__device__ __forceinline__ v8f wmma_f32(v2f a, v2f b, v8f c) {
  // D = A(16x4 f32) * B(4x16 f32) + C(16x16 f32)
  return __builtin_amdgcn_wmma_f32_16x16x4_f32(false, a, false, b, (short)0, c,
                                               false, false);
}

__device__ __forceinline__ v2f relu2(v2f v) {
  v2f r;
  r[0] = fmaxf(v[0], 0.f);
  r[1] = fmaxf(v[1], 0.f);
  return r;
}

// ---------------- degree helpers ----------------
__global__ void set_ones_kernel(float* __restrict__ p, int n) {
  int i = blockIdx.x * blockDim.x + threadIdx.x;
  if (i < n) p[i] = 1.0f;
}

__global__ void deg_count_kernel(float* __restrict__ deg,
                                 const int* __restrict__ dst, int E) {
  int i = blockIdx.x * blockDim.x + threadIdx.x;
  if (i < E) atomicAdd(&deg[dst[i]], 1.0f);
}

__global__ void rsqrt_kernel(float* __restrict__ p, int n) {
  int i = blockIdx.x * blockDim.x + threadIdx.x;
  if (i < n) p[i] = rsqrtf(p[i]);
}

// WT[n*K + k] = W[k*N + n]
__global__ void transpose_kernel(const float* __restrict__ W,
                                 float* __restrict__ WT, int K, int Nc) {
  int gid = blockIdx.x * blockDim.x + threadIdx.x;
  if (gid >= K * Nc) return;
  int k = gid / Nc;
  int n = gid - k * Nc;
  WT[(size_t)n * K + k] = W[gid];
}

// ---------------- dense node GEMM with fused epilogues -----------------------
// C[M][NT*16] = act(A[M][128]) @ W[128][NT*16]
//   IN_BIAS_RELU: A element -> relu(A + inBias[col])   (fuses prior finalize)
//   SELF_STORE  : also write S = C * dis[m]^2          (fuses self-loop init)
// One wave per 16-row M-tile; WT is the transposed weight [NT*16][128].
template <int NT, bool IN_BIAS_RELU, bool SELF_STORE>
__global__ __launch_bounds__(256) void gemm_node_kernel(
    const float* __restrict__ A, const float* __restrict__ WT,
    const float* __restrict__ inBias, const float* __restrict__ dis,
    float* __restrict__ C, float* __restrict__ S, int M) {
  int wid = (blockIdx.x * blockDim.x + threadIdx.x) >> 5;
  int lane = threadIdx.x & 31;
  int m0 = wid * 16;
  if (m0 >= M) return;                 // wave-uniform
  int row = lane & 15;
  int half = lane >> 4;
  int mrow = m0 + row;
  if (mrow >= M) mrow = M - 1;         // clamp loads; stores guarded below
  const float* ap = A + (size_t)mrow * 128 + 2 * half;

  v2f a[32];
#pragma unroll
  for (int ks = 0; ks < 32; ++ks) {
    v2f v = *(const v2f*)(ap + 4 * ks);
    if constexpr (IN_BIAS_RELU) {
      v2f bb = *(const v2f*)(inBias + 4 * ks + 2 * half);
      v = relu2(v + bb);
    }
    a[ks] = v;
  }

  v8f c[NT];
  const v8f z = {0.f, 0.f, 0.f, 0.f, 0.f, 0.f, 0.f, 0.f};
#pragma unroll
  for (int j = 0; j < NT; ++j) c[j] = z;

#pragma unroll
  for (int j = 0; j < NT; ++j) {
    const float* wp = WT + (size_t)(j * 16 + row) * 128 + 2 * half;
#pragma unroll
    for (int ks = 0; ks < 32; ++ks) {
      v2f b = *(const v2f*)(wp + 4 * ks);
      c[j] = wmma_f32(a[ks], b, c[j]);
    }
  }

  float dv[8];
  if constexpr (SELF_STORE) {
#pragma unroll
    for (int r = 0; r < 8; ++r) {
      int m = m0 + r + 8 * half;
      if (m >= M) m = M - 1;
      float t = dis[m];
      dv[r] = t * t;
    }
  }

  const int NCH = NT * 16;
#pragma unroll
  for (int j = 0; j < NT; ++j) {
#pragma unroll
    for (int r = 0; r < 8; ++r) {
      int m = m0 + r + 8 * half;
      if (m < M) {
        size_t idx = (size_t)m * NCH + j * 16 + row;
        float v = c[j][r];
        C[idx] = v;
        if constexpr (SELF_STORE) S[idx] = v * dv[r];
      }
    }
  }
}

// ---------------- GCN scatter: OUT[dst[e]][c] += H[src[e]][c]*dis_s*dis_d ----
template <int CH>
__global__ void scatter_kernel(const float* __restrict__ H,
                               const int* __restrict__ src,
                               const int* __restrict__ dst,
                               const float* __restrict__ dis,
                               float* __restrict__ OUT, int E) {
  const int PER = CH / 4;
  int gid = blockIdx.x * blockDim.x + threadIdx.x;
  if (gid >= E * PER) return;
  int e = gid / PER;
  int q = gid - e * PER;
  int s = src[e];
  int d = dst[e];
  float nrm = dis[s] * dis[d];
  float4 h = *(const float4*)(H + (size_t)s * CH + 4 * q);
  float* op = OUT + (size_t)d * CH + 4 * q;
  atomicAdd(op + 0, h.x * nrm);
  atomicAdd(op + 1, h.y * nrm);
  atomicAdd(op + 2, h.z * nrm);
  atomicAdd(op + 3, h.w * nrm);
}

// ---------------- per-edge MLP: relu([h_s||h_d] @ Wm1 + bm1) @ Wm2 + bm2 -----
// One wave per 16 edges. WmT = transposed Wm1: [128 n][128 k].
// H2raw is the pre-bias aggregated layer-2 output; b2 is fused into A loads.
__global__ __launch_bounds__(256) void edge_mlp_kernel(
    const float* __restrict__ H2raw, const float* __restrict__ b2,
    const int* __restrict__ src, const int* __restrict__ dst,
    const float* __restrict__ WmT, const float* __restrict__ bm1,
    const float* __restrict__ Wm2, const float* __restrict__ bm2,
    float* __restrict__ out, int E) {
  int wid = (blockIdx.x * blockDim.x + threadIdx.x) >> 5;
  int lane = threadIdx.x & 31;
  int e0 = wid * 16;
  if (e0 >= E) return;                 // wave-uniform
  int row = lane & 15;
  int half = lane >> 4;
  int e = e0 + row;
  if (e >= E) e = E - 1;               // clamp loads
  int s = src[e];
  int d = dst[e];
  const float* ps = H2raw + (size_t)s * 64 + 2 * half;
  const float* pd = H2raw + (size_t)d * 64 + 2 * half;

  // A tile: 16 edges x 128 feat (cols 0..63 = h2[src]+b2, 64..127 = h2[dst]+b2)
  v2f a[32];
#pragma unroll
  for (int ks = 0; ks < 16; ++ks) {
    v2f bb = *(const v2f*)(b2 + 4 * ks + 2 * half);
    a[ks]      = *(const v2f*)(ps + 4 * ks) + bb;
    a[16 + ks] = *(const v2f*)(pd + 4 * ks) + bb;
  }

  float acc[8];
#pragma unroll
  for (int r = 0; r < 8; ++r) acc[r] = 0.f;

  const v8f z = {0.f, 0.f, 0.f, 0.f, 0.f, 0.f, 0.f, 0.f};
#pragma unroll
  for (int j = 0; j < 8; ++j) {
    float b1v = bm1[j * 16 + row];   // hidden-channel bias (N = j*16 + lane%16)
    float w2v = Wm2[j * 16 + row];   // projection weight for this column
    const float* wp = WmT + (size_t)(j * 16 + row) * 128 + 2 * half;
    v8f c = z;
#pragma unroll
    for (int ks = 0; ks < 32; ++ks) {
      v2f b = *(const v2f*)(wp + 4 * ks);
      c = wmma_f32(a[ks], b, c);
    }
#pragma unroll
    for (int r = 0; r < 8; ++r) {
      float hdn = fmaxf(c[r] + b1v, 0.f);
      acc[r] = fmaf(hdn, w2v, acc[r]);
    }
  }

  // reduce over the 16 N-columns (lanes 0..15 hold rows 0..7, 16..31 rows 8..15)
#pragma unroll
  for (int off = 8; off >= 1; off >>= 1) {
#pragma unroll
    for (int r = 0; r < 8; ++r) acc[r] += __shfl_xor(acc[r], off, 16);
  }

  if (row == 0) {
    float bb2 = bm2[0];
    int base = e0 + 8 * half;
#pragma unroll
    for (int r = 0; r < 8; ++r)
      if (base + r < E) out[base + r] = acc[r] + bb2;
  }
}

// ---------------- launch ----------------
extern "C" void kernel_launch(void* const* d_in, const int* in_sizes, int n_in,
                              void* d_out, int out_size, void* d_ws,
                              size_t ws_size, hipStream_t stream) {
  const float* x   = (const float*)d_in[0];
  const int*   src = (const int*)d_in[1];
  const int*   dst = (const int*)d_in[2];
  const float* W1  = (const float*)d_in[4];
  const float* b1  = (const float*)d_in[5];
  const float* W2  = (const float*)d_in[6];
  const float* b2  = (const float*)d_in[7];
  const float* Wm1 = (const float*)d_in[8];
  const float* bm1 = (const float*)d_in[9];
  const float* Wm2 = (const float*)d_in[10];
  const float* bm2 = (const float*)d_in[11];
  float* out = (float*)d_out;

  const int Nn = in_sizes[0] / 128;
  const int E  = in_sizes[1];

  size_t off = 0;
  auto bump = [&](size_t bytes) -> float* {
    float* p = (float*)((char*)d_ws + off);
    off += (bytes + 255) & ~(size_t)255;
    return p;
  };
  float* dis  = bump((size_t)Nn * 4);
  float* W1T  = bump(128 * 128 * 4);
  float* W2T  = bump(128 * 64 * 4);
  float* WmT  = bump(128 * 128 * 4);
  float* buf1 = bump((size_t)Nn * 128 * 4);  // h1 / h2raw (gather side)
  float* buf2 = bump((size_t)Nn * 128 * 4);  // agg1 (self-init + scattered)
  float* buf3 = bump((size_t)Nn * 64 * 4);   // agg2 (self-init + scattered)

  const int B = 256;
  // degrees: deg = 1 + count(dst); dis = rsqrt(deg)
  set_ones_kernel<<<(Nn + B - 1) / B, B, 0, stream>>>(dis, Nn);
  deg_count_kernel<<<(E + B - 1) / B, B, 0, stream>>>(dis, dst, E);
  rsqrt_kernel<<<(Nn + B - 1) / B, B, 0, stream>>>(dis, Nn);

  // transposed weights for WMMA B-fragment float2 loads
  transpose_kernel<<<(128 * 128 + B - 1) / B, B, 0, stream>>>(W1, W1T, 128, 128);
  transpose_kernel<<<(128 * 64 + B - 1) / B, B, 0, stream>>>(W2, W2T, 128, 64);
  transpose_kernel<<<(128 * 128 + B - 1) / B, B, 0, stream>>>(Wm1, WmT, 128, 128);

  const int mtiles = (Nn + 15) / 16;
  const int gemm_blocks = (mtiles * 32 + B - 1) / B;

  // ---- layer 1: buf1 = x@W1 ; buf2 = buf1*dis^2 (fused self-loop init) ----
  gemm_node_kernel<8, false, true><<<gemm_blocks, B, 0, stream>>>(
      x, W1T, nullptr, dis, buf1, buf2, Nn);
  scatter_kernel<128><<<(E * 32 + B - 1) / B, B, 0, stream>>>(
      buf1, src, dst, dis, buf2, E);

  // ---- layer 2: A = relu(buf2 + b1) fused; buf1 = A@W2 ; buf3 = buf1*dis^2 --
  gemm_node_kernel<4, true, true><<<gemm_blocks, B, 0, stream>>>(
      buf2, W2T, b1, dis, buf1, buf3, Nn);
  scatter_kernel<64><<<(E * 16 + B - 1) / B, B, 0, stream>>>(
      buf1, src, dst, dis, buf3, E);

  // ---- edge MLP (b2 fused into gathered A fragments) ----
  const int etiles = (E + 15) / 16;
  edge_mlp_kernel<<<(etiles * 32 + B - 1) / B, B, 0, stream>>>(
      buf3, b2, src, dst, WmT, bm1, Wm2, bm2, out, E);
}